// HyperConnections_37056977830023
// MI455X (gfx1250) — compile-verified
//
#include <hip/hip_runtime.h>
#include <math.h>

// Problem constants (match reference)
#define MB   4        // batch
#define MS   4        // streams
#define MN   2048     // sequence length
#define MD   1024     // feature dim
#define MHC_ITERS 10
#define MHC_TAU   0.05f
#define BATCH 8       // j-tiles per software-pipeline stage (16 loads in flight)

typedef __attribute__((ext_vector_type(2))) float v2f;
typedef __attribute__((ext_vector_type(8))) float v8f;

// ---------------------------------------------------------------------------
// Kernel 1: tiny log-domain Sinkhorn + softmaxes -> combined 4x4 mix matrix M
//   M[s][t] = h_res[s][t] + beta[t] * h_pre[s]
// Written to d_ws (16 floats). One lane; ~2 KFLOP, negligible.
// ---------------------------------------------------------------------------
__device__ __forceinline__ float lse4(float a, float b, float c, float d) {
    float m = fmaxf(fmaxf(a, b), fmaxf(c, d));
    return m + logf(expf(a - m) + expf(b - m) + expf(c - m) + expf(d - m));
}

__global__ void mhc_coeffs_kernel(const float* __restrict__ Hres,
                                  const float* __restrict__ Hpre,
                                  const float* __restrict__ Hpost,
                                  float* __restrict__ M) {
    if (threadIdx.x != 0 || blockIdx.x != 0) return;

    float Z[MS][MS];
#pragma unroll
    for (int s = 0; s < MS; ++s)
#pragma unroll
        for (int t = 0; t < MS; ++t)
            Z[s][t] = Hres[s * MS + t] * (1.0f / MHC_TAU);

    float u[MS] = {0.f, 0.f, 0.f, 0.f};
    float v[MS] = {0.f, 0.f, 0.f, 0.f};
    for (int it = 0; it < MHC_ITERS; ++it) {
#pragma unroll
        for (int s = 0; s < MS; ++s)
            u[s] = -lse4(Z[s][0] + v[0], Z[s][1] + v[1],
                         Z[s][2] + v[2], Z[s][3] + v[3]);
#pragma unroll
        for (int t = 0; t < MS; ++t)
            v[t] = -lse4(Z[0][t] + u[0], Z[1][t] + u[1],
                         Z[2][t] + u[2], Z[3][t] + u[3]);
    }

    float hp[MS];
    {
        float m = fmaxf(fmaxf(Hpre[0], Hpre[1]), fmaxf(Hpre[2], Hpre[3]));
        float sum = 0.f;
#pragma unroll
        for (int i = 0; i < MS; ++i) { hp[i] = expf(Hpre[i] - m); sum += hp[i]; }
        float inv = 1.0f / sum;
#pragma unroll
        for (int i = 0; i < MS; ++i) hp[i] *= inv;
    }
    float bt[MS];
    {
        float m = fmaxf(fmaxf(Hpost[0], Hpost[1]), fmaxf(Hpost[2], Hpost[3]));
        float sum = 0.f;
#pragma unroll
        for (int i = 0; i < MS; ++i) { bt[i] = expf(Hpost[i] - m); sum += bt[i]; }
        float inv = 1.0f / sum;
#pragma unroll
        for (int i = 0; i < MS; ++i) bt[i] *= inv;
    }

#pragma unroll
    for (int s = 0; s < MS; ++s)
#pragma unroll
        for (int t = 0; t < MS; ++t)
            M[s * MS + t] = expf(Z[s][t] + u[s] + v[t]) + bt[t] * hp[s];
}

// ---------------------------------------------------------------------------
// Kernel 2: streaming stream-mix via V_WMMA_F32_16X16X4_F32.
//   D(16x16) = A(16x4 := M^T padded) x B(4x16 := input slab), fp32 throughout.
// One wave per (b,n) row; 64 WMMA tiles of 16 d-columns, software-pipelined
// in batches of 8: 16 NT loads -> 8 WMMAs -> one predicated NT store block.
// Layouts (ISA 7.12.2): A lane L holds A[L&15][K], K split {0,1}|{2,3} by lane
// half, VGPR = K LSB. B lane = column, same K split. D: VGPR r = row r in
// lanes 0-15 -> output stream r.
// ---------------------------------------------------------------------------
__global__ void __launch_bounds__(256)
mhc_mix_kernel(const float* __restrict__ R,   // residuals (B*S, N, D)
               const float* __restrict__ M,   // 16-float mix matrix (s-major)
               float* __restrict__ O) {       // output    (B*S, N, D)
    const int lane = threadIdx.x & 31;
    const int wid  = blockIdx.x * (blockDim.x >> 5) + (threadIdx.x >> 5);
    if (wid >= MB * MN) return;

    const int bIdx = wid >> 11;          // / MN   (MN = 2048)
    const int n    = wid & (MN - 1);
    const int col  = lane & 15;          // A row index / B,D column index
    const int hi   = lane >> 4;          // lane half -> K (stream) MSB
    const int sA   = hi ? 2 : 0;         // stream supplying VGPR0 of A and B
    const int sB   = hi ? 3 : 1;         // stream supplying VGPR1 of A and B

    // A = M^T (rows t<4 valid, rest zero). Two per-lane scalar loads, once.
    v2f A;
    A.x = (col < MS) ? M[sA * MS + col] : 0.0f;
    A.y = (col < MS) ? M[sB * MS + col] : 0.0f;

    const size_t plane  = (size_t)MN * MD;                  // stream plane
    const size_t baseBN = (size_t)bIdx * MS * plane + (size_t)n * MD;
    const float* pb0 = R + baseBN + (size_t)sA * plane + col;
    const float* pb1 = R + baseBN + (size_t)sB * plane + col;
    // Per-stream output base pointers -> all stores use small imm offsets.
    float* po0 = O + baseBN + col;
    float* po1 = po0 + plane;
    float* po2 = po1 + plane;
    float* po3 = po2 + plane;

    const bool active = (lane < 16);     // D rows 0..3 live in lanes 0-15

    for (int j = 0; j < MD / 16; j += BATCH) {
        // Stage 1: issue the whole batch of loads (distinct regs -> 16 in flight)
        float bx[BATCH], by[BATCH];
#pragma unroll
        for (int k = 0; k < BATCH; ++k) {
            bx[k] = __builtin_nontemporal_load(pb0 + (j + k) * 16);
            by[k] = __builtin_nontemporal_load(pb1 + (j + k) * 16);
        }
        // Stage 2: back-to-back WMMAs (EXEC all ones; independent D regs)
        v8f d[BATCH];
#pragma unroll
        for (int k = 0; k < BATCH; ++k) {
            v2f Bv; Bv.x = bx[k]; Bv.y = by[k];
            v8f c = {0.f, 0.f, 0.f, 0.f, 0.f, 0.f, 0.f, 0.f};
            // 8 args: (neg_a, A, neg_b, B, c_mod, C, reuse_a, reuse_b)
            d[k] = __builtin_amdgcn_wmma_f32_16x16x4_f32(
                false, A, false, Bv, (short)0, c, false, false);
        }
        // Stage 3: one predicated store block for the batch
        if (active) {
#pragma unroll
            for (int k = 0; k < BATCH; ++k) {
                const int o = (j + k) * 16;
                __builtin_nontemporal_store(d[k][0], po0 + o);
                __builtin_nontemporal_store(d[k][1], po1 + o);
                __builtin_nontemporal_store(d[k][2], po2 + o);
                __builtin_nontemporal_store(d[k][3], po3 + o);
            }
        }
    }
}

// ---------------------------------------------------------------------------
extern "C" void kernel_launch(void* const* d_in, const int* in_sizes, int n_in,
                              void* d_out, int out_size, void* d_ws, size_t ws_size,
                              hipStream_t stream) {
    const float* residuals = (const float*)d_in[0];   // (B*S, N, D) fp32
    const float* Hres      = (const float*)d_in[1];   // (S, S)
    const float* Hpre      = (const float*)d_in[2];   // (1, S)
    const float* Hpost     = (const float*)d_in[3];   // (1, S)
    float*       out       = (float*)d_out;
    float*       Mbuf      = (float*)d_ws;            // 16 floats scratch

    (void)in_sizes; (void)n_in; (void)out_size; (void)ws_size;

    mhc_coeffs_kernel<<<1, 32, 0, stream>>>(Hres, Hpre, Hpost, Mbuf);

    const int waves   = MB * MN;            // one wave per (b, n) row
    const int threads = 256;                // 8 waves / block
    const int blocks  = waves / (threads / 32);
    mhc_mix_kernel<<<blocks, threads, 0, stream>>>(residuals, Mbuf, out);
}